// cycnet_6665789243619
// MI455X (gfx1250) — compile-verified
//
#include <hip/hip_runtime.h>

typedef float v8f __attribute__((ext_vector_type(8)));
typedef float v2f __attribute__((ext_vector_type(2)));

#ifndef __has_builtin
#define __has_builtin(x) 0
#endif
#if __has_builtin(__builtin_amdgcn_wmma_f32_16x16x4_f32)
#define USE_WMMA 1
#else
#define USE_WMMA 0
#endif

#define Vn    1024
#define Ln    8
#define Bn    2048
#define En    8192
#define NIT   5
#define CLIPV 10.0f

// One workgroup per batch row; the whole 32KB message row lives in LDS.
// Permuted-storage invariant (idx_rc and idx_cr are inverse permutations):
//   lds_e[k] == e_msg[idx_rc[k]]   -> odd layer reads LINEARLY
//   lds_o[q] == odd[q]             -> odd layer writes LINEARLY
// The even layer both GATHERS (ev[l] = lds_o[idx_cr[e]]) and SCATTERS
// (lds_e[idx_cr[e]] = msg[e]) with the SAME index registers, so idx_rc is
// never needed and the permutation is applied exactly once per layer.
__global__ __launch_bounds__(256)
void cycnet_fused(const float* __restrict__ x,
                  const float* __restrict__ message,
                  const float* __restrict__ oddw_v,
                  const float* __restrict__ oddw_e,
                  const float* __restrict__ w_e_out,
                  const float* __restrict__ mask_e,
                  const int*   __restrict__ idx_cr,
                  float*       __restrict__ out)
{
  __shared__ float lds_e[En];
  __shared__ float lds_o[En];

  const int b    = blockIdx.x;
  const int tid  = threadIdx.x;
  const int lane = tid & 31;
  const int wave = tid >> 5;

  const float* xr = x + (size_t)b * Vn;
  const float* mr = message + (size_t)b * En;

  // ---- initial permuted load: lds_e[idx_cr[e]] = message[e] ----
  // async global->LDS with per-lane LDS destination == free scatter engine
  unsigned lds_e_base = (unsigned)(size_t)lds_e;
  for (int i = 0; i < En / 256; ++i) {
    int e = tid + (i << 8);
    unsigned la = lds_e_base + ((unsigned)idx_cr[e] << 2);
    unsigned long long ga = (unsigned long long)(size_t)(mr + e);
    asm volatile("global_load_async_to_lds_b32 %0, %1, off"
                 :: "v"(la), "v"(ga) : "memory");
  }
  asm volatile("s_wait_asynccnt 0x0" ::: "memory");
  __syncthreads();

  const int col   = lane & 15;          // D-matrix column this lane holds
  const int khalf = (lane >> 4) << 1;   // K offset for A/B operands: 0 or 2
  const int hi8   = (lane >> 4) << 3;   // row offset for D rows: 0 or 8

  for (int t = 0; t < NIT; ++t) {
    // per-iteration weights: Wcol[l] = (oddw_e[t] * mask)[l][col]
    float Wcol[Ln];
    float wvc = 0.f;
    #pragma unroll
    for (int l = 0; l < Ln; ++l) Wcol[l] = 0.f;
    if (col < Ln) {
      #pragma unroll
      for (int l = 0; l < Ln; ++l)
        Wcol[l] = oddw_e[t * Ln * Ln + l * Ln + col] * mask_e[l * Ln + col];
      wvc = oddw_v[t * Ln + col];
    }
#if USE_WMMA
    // B operand (4x16 f32, zero-padded cols 8..15):
    // VGPR0 = K=khalf row, VGPR1 = K=khalf+1 row; second WMMA covers K=4..7
    v2f bm0, bm1;
    bm0.x = Wcol[khalf];     bm0.y = Wcol[khalf + 1];
    bm1.x = Wcol[khalf + 4]; bm1.y = Wcol[khalf + 5];
#endif

    // ---------------- odd layer (linear read, linear write) ----------------
    // 8 waves x 8 tiles x 16 rows = 1024 rows
    const int arow0 = lane & 15;
    for (int tt = 0; tt < 8; ++tt) {
      const int vbase = wave * 128 + tt * 16;
      const int arow  = vbase + arow0;
#if USE_WMMA
      // A operand 16x4 f32: lane<16 -> K{0,1}, lane>=16 -> K{2,3}
      v2f a0 = *(const v2f*)&lds_e[arow * Ln + khalf];
      v2f a1 = *(const v2f*)&lds_e[arow * Ln + khalf + 4];
      v8f acc = {0.f,0.f,0.f,0.f,0.f,0.f,0.f,0.f};
      acc = __builtin_amdgcn_wmma_f32_16x16x4_f32(false, a0, false, bm0,
                                                  (short)0, acc, false, false);
      acc = __builtin_amdgcn_wmma_f32_16x16x4_f32(false, a1, false, bm1,
                                                  (short)0, acc, false, false);
#endif
      if (col < Ln) {
        #pragma unroll
        for (int j = 0; j < 8; ++j) {
          const int row = vbase + j + hi8;   // D VGPR j: rows j / j+8
#if USE_WMMA
          float eo = acc[j];
#else
          float eo = 0.f;
          #pragma unroll
          for (int l = 0; l < Ln; ++l)
            eo += lds_e[row * Ln + l] * Wcol[l];
#endif
          float val = eo + xr[row] * wvc;
          val = fminf(fmaxf(val, -CLIPV), CLIPV);
          float ex = __expf(val);
          // tanh(val/2) = (e^val - 1) / (e^val + 1), fast rcp instead of div
          float o = (ex - 1.f) * __builtin_amdgcn_rcpf(ex + 1.f);
          lds_o[row * Ln + col] = o;          // LINEAR store
        }
      }
    }
    __syncthreads();

    // ---------------- even layer (gather + scatter via idx_cr) ------------
    for (int k = 0; k < 4; ++k) {
      const int v = tid + (k << 8);
      const int4 i0 = *(const int4*)(idx_cr + (size_t)v * Ln);
      const int4 i1 = *(const int4*)(idx_cr + (size_t)v * Ln + 4);
      const int idx[Ln] = { i0.x, i0.y, i0.z, i0.w, i1.x, i1.y, i1.z, i1.w };
      float ev[Ln];
      #pragma unroll
      for (int l = 0; l < Ln; ++l) {
        float u = lds_o[idx[l]];              // gather
        ev[l] = (u == 0.f) ? 1.f : u;         // replace exact zeros by 1
      }
      // leave-one-out product via prefix x suffix (no divisions)
      float pre[Ln], suf[Ln];
      pre[0] = 1.f;
      #pragma unroll
      for (int l = 1; l < Ln; ++l) pre[l] = pre[l - 1] * ev[l - 1];
      suf[Ln - 1] = 1.f;
      #pragma unroll
      for (int l = Ln - 2; l >= 0; --l) suf[l] = suf[l + 1] * ev[l + 1];
      #pragma unroll
      for (int l = 0; l < Ln; ++l) {
        float u = pre[l] * suf[l];
        if (t == 0) u = fminf(fmaxf(u, -CLIPV), CLIPV);
        // 2*atanh(u) = log((1+u)/(1-u)), fast rcp instead of div
        float m = __logf((1.f + u) * __builtin_amdgcn_rcpf(1.f - u));
        lds_e[idx[l]] = m;                    // scatter (same indices)
      }
    }
    __syncthreads();
  }

  // ---------------- output: x + ep @ w_e_out (ep is linear in lds_e) ------
  float w8[Ln];
  #pragma unroll
  for (int l = 0; l < Ln; ++l) w8[l] = w_e_out[l];
  for (int k = 0; k < 4; ++k) {
    const int v = tid + (k << 8);
    float acc = xr[v];
    #pragma unroll
    for (int l = 0; l < Ln; ++l) acc += lds_e[v * Ln + l] * w8[l];
    out[(size_t)b * Vn + v] = acc;
  }
}

extern "C" void kernel_launch(void* const* d_in, const int* in_sizes, int n_in,
                              void* d_out, int out_size, void* d_ws, size_t ws_size,
                              hipStream_t stream) {
  (void)in_sizes; (void)n_in; (void)out_size; (void)d_ws; (void)ws_size;
  const float* x        = (const float*)d_in[0];
  const float* message  = (const float*)d_in[1];
  const float* oddw_v   = (const float*)d_in[2];
  const float* oddw_e   = (const float*)d_in[3];
  const float* w_e_out  = (const float*)d_in[4];
  const float* mask_e   = (const float*)d_in[5];
  const int*   idx_cr   = (const int*)d_in[7];   // idx_rc (d_in[6]) not needed
  float*       out      = (float*)d_out;

  cycnet_fused<<<dim3(Bn), dim3(256), 0, stream>>>(
      x, message, oddw_v, oddw_e, w_e_out, mask_e, idx_cr, out);
}